// Tensor3D_15719580304449
// MI455X (gfx1250) — compile-verified
//
#include <hip/hip_runtime.h>

namespace {

constexpr float kBounds = 1.6f;
constexpr int   kFeat   = 32;
constexpr int   kGrids  = 9;

// 8-byte texel pair with 4-byte alignment (pairs are only DWORD-aligned in the grid).
struct __attribute__((packed, aligned(4))) F2 { float lo, hi; };

struct Samp {
  const F2* r0;   // row ys, starting at xs; row ys+1 is at constant +R*4 bytes
  float wx, wy;
};

// Edge handling without selects: xs = min(floor(x), R-2), wx = x - xs.
// Interior: identical to the reference (wx = frac(x)). At the clamped edge
// x == R-1: wx == 1.0 exactly, so the lerp returns the .hi texel (== G[R-1]).
// Same for y, which also removes the y1 = min(y0+1, R-1) clamp.
template <int R>
__device__ __forceinline__ Samp prep(const float* __restrict__ g,
                                     float cx, float cy, int lane) {
  const float mx = (float)(R - 1);
  float x = fminf(fmaxf((cx + 1.0f) * (0.5f * mx), 0.0f), mx);
  float y = fminf(fmaxf((cy + 1.0f) * (0.5f * mx), 0.0f), mx);
  int xs = min((int)x, R - 2);
  int ys = min((int)y, R - 2);
  Samp s;
  s.r0 = (const F2*)(g + lane * (R * R) + ys * R + xs);
  s.wx = x - (float)xs;
  s.wy = y - (float)ys;
  return s;
}

__device__ __forceinline__ float blend(const Samp& s, F2 a, F2 b) {
  float t0 = a.lo + s.wx * (a.hi - a.lo);
  float t1 = b.lo + s.wx * (b.hi - b.lo);
  return t0 + s.wy * (t1 - t0);
}

}  // namespace

__global__ __launch_bounds__(256) void triplane_sample_kernel(
    const float* __restrict__ pts,
    const float* __restrict__ g00, const float* __restrict__ g01, const float* __restrict__ g02,
    const float* __restrict__ g10, const float* __restrict__ g11, const float* __restrict__ g12,
    const float* __restrict__ g20, const float* __restrict__ g21, const float* __restrict__ g22,
    float* __restrict__ out, int n) {
  int wave = (int)((blockIdx.x * blockDim.x + threadIdx.x) >> 5);  // one wave32 per point
  int lane = (int)(threadIdx.x & 31);                              // lane = feature channel
  if (wave >= n) return;

  // Wave-uniform point load + normalization: p = (pt - B) * (2 / (-2B)) - 1
  float p0 = pts[wave * 3 + 0];
  float p1 = pts[wave * 3 + 1];
  float p2 = pts[wave * 3 + 2];
  constexpr float kScale = 2.0f / (-2.0f * kBounds);
  float q0 = (p0 - kBounds) * kScale - 1.0f;
  float q1 = (p1 - kBounds) * kScale - 1.0f;
  float q2 = (p2 - kBounds) * kScale - 1.0f;

  // planes: idx0 -> (x=q1, y=q2), idx1 -> (x=q0, y=q2), idx2 -> (x=q0, y=q1)
  Samp s[kGrids];
  s[0] = prep<128>(g00, q1, q2, lane);
  s[1] = prep<128>(g01, q0, q2, lane);
  s[2] = prep<128>(g02, q0, q1, lane);
  s[3] = prep<256>(g10, q1, q2, lane);
  s[4] = prep<256>(g11, q0, q2, lane);
  s[5] = prep<256>(g12, q0, q1, lane);
  s[6] = prep<512>(g20, q1, q2, lane);
  s[7] = prep<512>(g21, q0, q2, lane);
  s[8] = prep<512>(g22, q0, q1, lane);

  // All 18 paired loads issued back-to-back for maximum MLP. Row 1 sits at a
  // compile-time byte offset (R*4) from row 0, folded into the load IOFFSET.
  F2 a[kGrids], b[kGrids];
#pragma unroll
  for (int i = 0; i < kGrids; ++i) {
    const int halfR = (i < 3) ? (128 / 2) : (i < 6) ? (256 / 2) : (512 / 2);
    a[i] = s[i].r0[0];
    b[i] = s[i].r0[halfR];  // +R F2? no: F2 is 8B, +R*4 bytes == +R/2 F2 elems
  }

  // Nine coalesced 128B row stores per wave; non-temporal so the 345MB output
  // stream does not evict the L2-resident (126MB) grids.
  float* o = out + (size_t)wave * (kGrids * kFeat) + lane;
#pragma unroll
  for (int i = 0; i < kGrids; ++i) {
    __builtin_nontemporal_store(blend(s[i], a[i], b[i]), o + i * kFeat);
  }
}

extern "C" void kernel_launch(void* const* d_in, const int* in_sizes, int n_in,
                              void* d_out, int out_size, void* d_ws, size_t ws_size,
                              hipStream_t stream) {
  const float* pts = (const float*)d_in[0];
  const float* g00 = (const float*)d_in[1];
  const float* g01 = (const float*)d_in[2];
  const float* g02 = (const float*)d_in[3];
  const float* g10 = (const float*)d_in[4];
  const float* g11 = (const float*)d_in[5];
  const float* g12 = (const float*)d_in[6];
  const float* g20 = (const float*)d_in[7];
  const float* g21 = (const float*)d_in[8];
  const float* g22 = (const float*)d_in[9];

  int n = in_sizes[0] / 3;                 // number of points
  int waves_per_block = 256 / 32;          // wave32 -> 8 waves per 256-thread block
  int blocks = (n + waves_per_block - 1) / waves_per_block;

  triplane_sample_kernel<<<blocks, 256, 0, stream>>>(
      pts, g00, g01, g02, g10, g11, g12, g20, g21, g22, (float*)d_out, n);
}